// CenterLoss_11828339933241
// MI455X (gfx1250) — compile-verified
//
#include <hip/hip_runtime.h>

typedef __attribute__((ext_vector_type(2))) float v2f;
typedef __attribute__((ext_vector_type(8))) float v8f;

#define BATCH 16384
#define FDIM  64
#define EPSN  1e-12f

// Each wave (32 lanes) handles one 16-row tile.
// Lane l: row r = tile*16 + (l&15), k-phase koff = 2*(l>>4).
// A-matrix (16x4 f32) lane layout: lanes 0-15 hold K=k0,k0+1 of row M=lane,
// lanes 16-31 hold K=k0+2,k0+3 of row M=lane-16  -> float2 at (r, k0+koff).
// B-matrix (4x16 f32) lane layout mirrors it with N=lane&15 -> the gathered
// center row uses the *same* per-lane addressing. 16 WMMA_F32_16X16X4_F32
// chunks accumulate D = Xn * G^T; we consume only the diagonal.
__global__ __launch_bounds__(256) void center_loss_main(
    const float* __restrict__ x,
    const int*   __restrict__ labels,
    const float* __restrict__ centers,
    float*       __restrict__ partial)
{
    __shared__ float wsum[8];
    const int lane = threadIdx.x & 31;
    const int wave = threadIdx.x >> 5;
    const int tile = blockIdx.x * 8 + wave;          // 0..1023
    const int r    = tile * 16 + (lane & 15);        // global row, < 16384
    const int koff = (lane >> 4) * 2;                // 0 or 2

    const int lab = labels[r];
    const float* __restrict__ xrow = x + (size_t)r * FDIM;
    const float* __restrict__ crow = centers + (size_t)lab * FDIM;

    v2f xv[16];
    v2f gv[16];
    float ss = 0.0f;   // partial ||x_r||^2 (this half-lane's K subset)
    float tt = 0.0f;   // partial ||c_{l_r}||^2
#pragma unroll
    for (int t = 0; t < 16; ++t) {
        xv[t] = *reinterpret_cast<const v2f*>(xrow + koff + 4 * t);
        gv[t] = *reinterpret_cast<const v2f*>(crow + koff + 4 * t);
        ss += xv[t].x * xv[t].x + xv[t].y * xv[t].y;
        tt += gv[t].x * gv[t].x + gv[t].y * gv[t].y;
    }
    // Combine the two half-lanes that share a row (K halves 0,1 mod4 / 2,3 mod4).
    ss += __shfl_xor(ss, 16, 32);
    tt += __shfl_xor(tt, 16, 32);

    const float inv = 1.0f / fmaxf(__builtin_sqrtf(ss), EPSN); // F.normalize eps
    const float s   = ss * inv * inv;                          // ||nor_x_r||^2
#pragma unroll
    for (int t = 0; t < 16; ++t) {
        xv[t].x *= inv;
        xv[t].y *= inv;
    }

    // D(16x16) += A(16x4) * B(4x16), 16 chunks over K=64.
    v8f c = {};
#pragma unroll
    for (int t = 0; t < 16; ++t) {
        c = __builtin_amdgcn_wmma_f32_16x16x4_f32(
                /*neg_a=*/false, xv[t], /*neg_b=*/false, gv[t],
                /*c_mod=*/(short)0, c, /*reuse_a=*/false, /*reuse_b=*/false);
    }

    // Diagonal of C: element (m,m): for m=0..7 -> VGPR m, lane m;
    // for m=8..15 -> VGPR m-8, lane 16+m. Lanes 0-7 and 24-31 each own one.
    float contrib = 0.0f;
    if (lane < 8 || lane >= 24) {
        const int idx = lane & 7;
        const float d = c[idx];
        contrib = s + tt - 2.0f * d;   // s,tt on this lane belong to row r
    }

    // Deterministic wave32 reduction.
#pragma unroll
    for (int off = 16; off >= 1; off >>= 1)
        contrib += __shfl_xor(contrib, off, 32);

    if (lane == 0) wsum[wave] = contrib;
    __syncthreads();
    if (threadIdx.x == 0) {
        float bs = 0.0f;
#pragma unroll
        for (int w = 0; w < 8; ++w) bs += wsum[w];  // fixed order
        partial[blockIdx.x] = bs;
    }
}

__global__ __launch_bounds__(128) void center_loss_final(
    const float* __restrict__ partial, float* __restrict__ out)
{
    __shared__ float sm[128];
    const int t = threadIdx.x;
    sm[t] = partial[t];
    __syncthreads();
#pragma unroll
    for (int s = 64; s >= 1; s >>= 1) {
        if (t < s) sm[t] += sm[t + s];
        __syncthreads();
    }
    if (t == 0) out[0] = sm[0] * (1.0f / (float)BATCH);
}

extern "C" void kernel_launch(void* const* d_in, const int* in_sizes, int n_in,
                              void* d_out, int out_size, void* d_ws, size_t ws_size,
                              hipStream_t stream)
{
    (void)in_sizes; (void)n_in; (void)out_size; (void)ws_size;
    const float* x       = (const float*)d_in[0];
    const int*   labels  = (const int*)d_in[1];
    const float* centers = (const float*)d_in[2];
    float* partial = (float*)d_ws;      // 128 floats, fully rewritten each call
    float* out     = (float*)d_out;

    // 16384 rows / (8 waves * 16 rows) = 128 blocks, exact cover (EXEC all-1s).
    center_loss_main<<<128, 256, 0, stream>>>(x, labels, centers, partial);
    center_loss_final<<<1, 128, 0, stream>>>(partial, out);
}